// GATLayer_257698038185
// MI455X (gfx1250) — compile-verified
//
#include <hip/hip_runtime.h>

#define N_NODES   50000
#define N_EDGES   800000
#define N_FEAT    128
#define N_HIDDEN  64
#define LRELU_A   0.05f
#define WT_STRIDE 130   // padded LDS stride (even -> 8B-aligned b64 ds loads)

typedef float v2f __attribute__((ext_vector_type(2)));
typedef float v8f __attribute__((ext_vector_type(8)));

// ---------------------------------------------------------------------------
// Kernel 0: fold a_w through W_fc:  wv[0..127] = W_fc @ a_w[:H],
//                                   wv[128..255] = W_fc @ a_w[H:]
// (s_dst = z@a1 = x@(W@a1), so the GEMM can emit scores directly.)
// ---------------------------------------------------------------------------
__global__ void gat_wv_kernel(const float* __restrict__ W,
                              const float* __restrict__ a_w,
                              float* __restrict__ wv) {
    int k = blockIdx.x * blockDim.x + threadIdx.x;
    if (k >= N_FEAT) return;
    const float* __restrict__ wr = W + (long)k * N_HIDDEN;
    float s1 = 0.0f, s2 = 0.0f;
    #pragma unroll 8
    for (int h = 0; h < N_HIDDEN; ++h) {
        float w = wr[h];
        s1 += w * a_w[h];
        s2 += w * a_w[N_HIDDEN + h];
    }
    wv[k]          = s1;
    wv[N_FEAT + k] = s2;
}

// ---------------------------------------------------------------------------
// Kernel 1: zero the accumulation buffers (out region of d_out, h_sum in ws)
// ---------------------------------------------------------------------------
__global__ void gat_zero_kernel(float* __restrict__ out_accum,
                                float* __restrict__ h_sum) {
    int i = blockIdx.x * blockDim.x + threadIdx.x;
    if (i < N_NODES * N_HIDDEN) out_accum[i] = 0.0f;
    if (i < N_NODES)            h_sum[i]     = 0.0f;
}

// ---------------------------------------------------------------------------
// Kernel 2: z = x @ W_fc via V_WMMA_F32_16X16X4_F32, with W staged transposed
// in LDS (one ds_load_b64 per B fragment) and a 5th WMMA accumulator whose
// columns 0/1 produce s_dst/s_src directly. The score-column fragment is
// loaded unconditionally (always in-bounds) and masked by a 0/1 multiply to
// avoid per-iteration exec-mask divergence.
// A(16x4): row = lane&15, K = 2*(lane>>4)+{0,1}
// B(4x16): col = lane&15, same K striping
// C/D(16x16): vgpr v, lane L -> row v + 8*(L>>4), col L&15
// ---------------------------------------------------------------------------
__global__ __launch_bounds__(128)
void gat_gemm_wmma_kernel(const float* __restrict__ x,
                          const float* __restrict__ W,
                          const float* __restrict__ wv,
                          float* __restrict__ z,
                          float* __restrict__ s_dst,
                          float* __restrict__ s_src) {
    __shared__ float wt[N_HIDDEN * WT_STRIDE];     // wt[n*130 + k] = W[k][n]
    __shared__ float wvl[2 * N_FEAT];              // staged score vectors

    const int tid = threadIdx.x;
    // cooperative staging: coalesced global reads, transposed LDS writes
    for (int idx = tid; idx < N_FEAT * N_HIDDEN; idx += 128) {
        int k = idx >> 6;                          // /N_HIDDEN
        int n = idx & (N_HIDDEN - 1);
        wt[n * WT_STRIDE + k] = W[idx];
    }
    wvl[tid]       = wv[tid];
    wvl[tid + 128] = wv[tid + 128];
    __syncthreads();

    const int lane = tid & 31;
    const int wave = tid >> 5;
    const int tile = blockIdx.x * 4 + wave;        // 16-row tile (3125 total)
    if (tile * 16 >= N_NODES) return;

    const int row   = tile * 16 + (lane & 15);
    const int khalf = (lane >> 4) * 2;             // 0 or 2
    const int ncol  = lane & 15;
    const float smask = (ncol < 2) ? 1.0f : 0.0f;  // score columns live in n=0,1

    v8f acc0 = {}, acc1 = {}, acc2 = {}, acc3 = {}, acc4 = {};
    const float* __restrict__ xr  = x + (long)row * N_FEAT;
    const float* wvp = wvl + (ncol & 1) * N_FEAT;  // always in-bounds for all lanes

    #pragma unroll 4
    for (int k0 = 0; k0 < N_FEAT; k0 += 4) {
        __builtin_prefetch(xr + k0 + 16, 0, 3);
        v2f a;
        a.x = xr[k0 + khalf];
        a.y = xr[k0 + khalf + 1];
        const float* wtk = wt + k0 + khalf;        // + n*WT_STRIDE (LDS)
        v2f b0 = *(const v2f*)(wtk + (ncol)      * WT_STRIDE);
        v2f b1 = *(const v2f*)(wtk + (16 + ncol) * WT_STRIDE);
        v2f b2 = *(const v2f*)(wtk + (32 + ncol) * WT_STRIDE);
        v2f b3 = *(const v2f*)(wtk + (48 + ncol) * WT_STRIDE);
        v2f bs = *(const v2f*)(wvp + k0 + khalf);  // unconditional ds_load_b64
        v2f b4;
        b4.x = bs.x * smask;
        b4.y = bs.y * smask;
        acc0 = __builtin_amdgcn_wmma_f32_16x16x4_f32(false, a, false, b0, (short)0, acc0, false, false);
        acc1 = __builtin_amdgcn_wmma_f32_16x16x4_f32(false, a, false, b1, (short)0, acc1, false, false);
        acc2 = __builtin_amdgcn_wmma_f32_16x16x4_f32(false, a, false, b2, (short)0, acc2, false, false);
        acc3 = __builtin_amdgcn_wmma_f32_16x16x4_f32(false, a, false, b3, (short)0, acc3, false, false);
        acc4 = __builtin_amdgcn_wmma_f32_16x16x4_f32(false, a, false, b4, (short)0, acc4, false, false);
    }

    const int rbase = tile * 16 + 8 * (lane >> 4);
    #pragma unroll
    for (int v = 0; v < 8; ++v) {
        float* __restrict__ zr = z + (long)(rbase + v) * N_HIDDEN + ncol;
        zr[0]  = acc0[v];
        zr[16] = acc1[v];
        zr[32] = acc2[v];
        zr[48] = acc3[v];
    }
    if (ncol == 0) {
        #pragma unroll
        for (int v = 0; v < 8; ++v) s_dst[rbase + v] = acc4[v];
    } else if (ncol == 1) {
        #pragma unroll
        for (int v = 0; v < 8; ++v) s_src[rbase + v] = acc4[v];
    }
}

// ---------------------------------------------------------------------------
// Kernel 3: per-edge h = exp(leakyrelu(...)); segment-sum into h_sum via atomics.
// h is staged in the alpha slice of d_out (normalized in-place by kernel 4).
// ---------------------------------------------------------------------------
__global__ void gat_edge_kernel(const int* __restrict__ ei,
                                const float* __restrict__ s_dst,
                                const float* __restrict__ s_src,
                                const float* __restrict__ a_b,
                                float* __restrict__ h_out,
                                float* __restrict__ h_sum) {
    int e = blockIdx.x * blockDim.x + threadIdx.x;
    if (e >= N_EDGES) return;
    int d = ei[e];
    int s = ei[N_EDGES + e];
    float h = s_dst[d] + s_src[s] + a_b[0];
    h = (h >= 0.0f) ? h : LRELU_A * h;
    h = __expf(h);
    h_out[e] = h;
    atomicAdd(&h_sum[d], h);
}

// ---------------------------------------------------------------------------
// Kernel 4: wave-per-edge: alpha = h / h_sum[dst] (written back in-place),
// then out[dst] += alpha * z[src] with 64-wide fp32 scatter atomics.
// z/out are L2-resident (12.8 MB each vs 192 MB L2).
// ---------------------------------------------------------------------------
__global__ __launch_bounds__(256)
void gat_scatter_kernel(const int* __restrict__ ei,
                        const float* __restrict__ z,
                        const float* __restrict__ h_sum,
                        float* __restrict__ alpha_io,
                        float* __restrict__ out) {
    const int lane = threadIdx.x & 31;
    const int e = blockIdx.x * (blockDim.x >> 5) + (threadIdx.x >> 5);
    if (e >= N_EDGES) return;
    const int d = ei[e];
    const int s = ei[N_EDGES + e];
    const float alpha = alpha_io[e] / h_sum[d];
    if (lane == 0) alpha_io[e] = alpha;
    const float* __restrict__ zs = z + (long)s * N_HIDDEN;
    float* __restrict__ od = out + (long)d * N_HIDDEN;
    atomicAdd(&od[lane],      alpha * zs[lane]);
    atomicAdd(&od[lane + 32], alpha * zs[lane + 32]);
}

// ---------------------------------------------------------------------------
extern "C" void kernel_launch(void* const* d_in, const int* in_sizes, int n_in,
                              void* d_out, int out_size, void* d_ws, size_t ws_size,
                              hipStream_t stream) {
    const float* x    = (const float*)d_in[0];   // [N_NODES, N_FEAT]
    const int*   ei   = (const int*)  d_in[1];   // [2, N_EDGES] (dst row 0, src row 1)
    const float* W_fc = (const float*)d_in[2];   // [N_FEAT, N_HIDDEN]
    const float* a_w  = (const float*)d_in[3];   // [2*N_HIDDEN, 1]
    const float* a_b  = (const float*)d_in[4];   // [1]

    float* out   = (float*)d_out;                        // [N_NODES, N_HIDDEN]
    float* alpha = out + (long)N_NODES * N_HIDDEN;       // [N_EDGES]

    // workspace layout (floats)
    float* z     = (float*)d_ws;                         // N_NODES*N_HIDDEN
    float* s_dst = z + (long)N_NODES * N_HIDDEN;         // N_NODES
    float* s_src = s_dst + N_NODES;                      // N_NODES
    float* h_sum = s_src + N_NODES;                      // N_NODES
    float* wv    = h_sum + N_NODES;                      // 2*N_FEAT

    // 0) fold a_w through W_fc
    gat_wv_kernel<<<1, 128, 0, stream>>>(W_fc, a_w, wv);
    // 1) zero accumulators
    {
        int total = N_NODES * N_HIDDEN;
        gat_zero_kernel<<<(total + 255) / 256, 256, 0, stream>>>(out, h_sum);
    }
    // 2) z = x @ W_fc (+ fused scores), WMMA fp32, 4 wave-tiles per block
    {
        int tiles  = N_NODES / 16;                 // 3125
        int blocks = (tiles + 3) / 4;              // 782
        gat_gemm_wmma_kernel<<<blocks, 128, 0, stream>>>(x, W_fc, wv, z, s_dst, s_src);
    }
    // 3) per-edge exp + segment sum (h staged in alpha slice)
    gat_edge_kernel<<<(N_EDGES + 255) / 256, 256, 0, stream>>>(ei, s_dst, s_src, a_b, alpha, h_sum);
    // 4) normalize + scatter (one wave per edge, 8 waves per block)
    {
        int wavesPerBlock = 256 / 32;
        int blocks = (N_EDGES + wavesPerBlock - 1) / wavesPerBlock;
        gat_scatter_kernel<<<blocks, 256, 0, stream>>>(ei, z, h_sum, alpha, out);
    }
}